// GCN_12567074308662
// MI455X (gfx1250) — compile-verified
//
#include <hip/hip_runtime.h>
#include <stdint.h>

// ---------------------------------------------------------------------------
// 2-layer GCN for MI455X (gfx1250), fp32 end-to-end.
// GEMMs use native fp32 WMMA: V_WMMA_F32_16X16X4_F32 (exact fp32 semantics).
// Edge aggregation: float4 gather + global_atomic_add_f32 scatter (L2-resident).
// ---------------------------------------------------------------------------

typedef __attribute__((ext_vector_type(2))) float v2f;
typedef __attribute__((ext_vector_type(8))) float v8f;

#define N_NODES 50000   // == 3125 * 16, so M-tiles need no tail handling
#define N_EDGES 800000
#define IN_C    128
#define HID_C   128
#define OUT_C   64

// H[M x NC] = op(X[M x 128]) @ W[128 x NC]   (op = ReLU if RELU_IN)
// One wave computes one 16x16 tile of H; 4 waves per block along N-tiles.
// WMMA f32 16x16x4 layouts (ISA 7.12.2):
//   A 16x4 : lanes 0-15 -> K={k,k+1}, lanes 16-31 -> K={k+2,k+3}, M = lane&15
//   B 4x16 : vgpr0 -> K=k(+2 for hi lanes), vgpr1 -> K=k+1(+3), N = lane&15
//   C/D    : vgpr r -> M = r + 8*(lane>>4), N = lane&15
template <int NC, bool RELU_IN>
__global__ __launch_bounds__(128)
void gemm_f32_wmma(const float* __restrict__ X,
                   const float* __restrict__ W,
                   float* __restrict__ H) {
    const int lane = threadIdx.x;        // 0..31 (wave32)
    const int sub  = lane & 15;
    const int half = lane >> 4;
    const int tileM = blockIdx.x;
    const int tileN = blockIdx.y * blockDim.y + threadIdx.y;

    const int row = tileM * 16 + sub;    // A-row this lane feeds
    const int col = tileN * 16 + sub;    // B-col this lane feeds

    const float* __restrict__ xrow = X + (size_t)row * 128;

    v8f acc = {};
#pragma unroll 8
    for (int k = 0; k < 128; k += 4) {
        const int ka = k + half * 2;
        v2f a;
        a.x = xrow[ka + 0];
        a.y = xrow[ka + 1];
        if (RELU_IN) {
            a.x = a.x > 0.0f ? a.x : 0.0f;
            a.y = a.y > 0.0f ? a.y : 0.0f;
        }
        v2f b;
        b.x = W[(size_t)(ka + 0) * NC + col];
        b.y = W[(size_t)(ka + 1) * NC + col];
        // (neg_a, A, neg_b, B, c_mod, C, reuse_a, reuse_b)
        acc = __builtin_amdgcn_wmma_f32_16x16x4_f32(
            false, a, false, b, (short)0, acc, false, false);
    }

    float* __restrict__ hout =
        H + (size_t)(tileM * 16 + half * 8) * NC + (size_t)tileN * 16 + sub;
#pragma unroll
    for (int r = 0; r < 8; ++r)
        hout[(size_t)r * NC] = acc[r];
}

// agg[i][c] = bias[c]  (seed scatter target with bias -> bias-add is free)
__global__ void init_bias_kernel(float* __restrict__ agg,
                                 const float* __restrict__ bias,
                                 int total, int cmask) {
    int i = blockIdx.x * blockDim.x + threadIdx.x;
    if (i < total) agg[i] = bias[i & cmask];
}

// agg[dst[e]] += w[e] * h[src[e]]   (one thread per (edge, 4-channel group))
template <int C, bool WEIGHTED>
__global__ void edge_scatter_kernel(const float* __restrict__ h,
                                    const long long* __restrict__ src,
                                    const long long* __restrict__ dst,
                                    const float* __restrict__ ew,
                                    float* __restrict__ agg) {
    constexpr int G  = C / 4;                 // float4 groups per edge
    constexpr int GS = (C == 128) ? 5 : 4;    // log2(G)
    int gid = blockIdx.x * blockDim.x + threadIdx.x;
    if (gid >= N_EDGES * G) return;
    const int e = gid >> GS;
    const int g = gid & (G - 1);
    const int s = (int)src[e];
    const int d = (int)dst[e];
    const float w = WEIGHTED ? ew[e] : 1.0f;
    const float4 v = *(const float4*)(h + (size_t)s * C + g * 4);
    float* __restrict__ o = agg + (size_t)d * C + g * 4;
    atomicAdd(o + 0, v.x * w);
    atomicAdd(o + 1, v.y * w);
    atomicAdd(o + 2, v.z * w);
    atomicAdd(o + 3, v.w * w);
}

extern "C" void kernel_launch(void* const* d_in, const int* in_sizes, int n_in,
                              void* d_out, int out_size, void* d_ws, size_t ws_size,
                              hipStream_t stream) {
    (void)in_sizes; (void)n_in; (void)out_size; (void)ws_size;

    const float*     x  = (const float*)d_in[0];
    const long long* ei = (const long long*)d_in[1];   // int64 edge_index [2, E]
    const float*     ew = (const float*)d_in[2];
    const float*     W1 = (const float*)d_in[3];
    const float*     b1 = (const float*)d_in[4];
    const float*     W2 = (const float*)d_in[5];
    const float*     b2 = (const float*)d_in[6];
    float* out = (float*)d_out;                        // 50000 x 64, doubles as agg2

    const long long* src = ei;            // row 0
    const long long* dst = ei + N_EDGES;  // row 1

    float* h1   = (float*)d_ws;                         // 50000 x 128
    float* agg1 = h1 + (size_t)N_NODES * HID_C;         // 50000 x 128
    float* h2   = h1;                                   // reuse: h1 dead after scatter 1

    // 1) h1 = x @ W1
    gemm_f32_wmma<HID_C, false>
        <<<dim3(N_NODES / 16, HID_C / 64), dim3(32, 4), 0, stream>>>(x, W1, h1);

    // 2) agg1 = b1 (broadcast)
    {
        const int total = N_NODES * HID_C;
        init_bias_kernel<<<(total + 255) / 256, 256, 0, stream>>>(agg1, b1, total, HID_C - 1);
    }

    // 3) agg1 += scatter(w * h1[src] -> dst)
    {
        const int total = N_EDGES * (HID_C / 4);
        edge_scatter_kernel<HID_C, true>
            <<<(total + 255) / 256, 256, 0, stream>>>(h1, src, dst, ew, agg1);
    }

    // 4) h2 = relu(agg1) @ W2    (ReLU fused into A-loads)
    gemm_f32_wmma<OUT_C, true>
        <<<dim3(N_NODES / 16, OUT_C / 64), dim3(32, 4), 0, stream>>>(agg1, W2, h2);

    // 5) out = b2 (broadcast)
    {
        const int total = N_NODES * OUT_C;
        init_bias_kernel<<<(total + 255) / 256, 256, 0, stream>>>(out, b2, total, OUT_C - 1);
    }

    // 6) out += scatter(h2[src] -> dst)   (second conv has no edge weight)
    {
        const int total = N_EDGES * (OUT_C / 4);
        edge_scatter_kernel<OUT_C, false>
            <<<(total + 255) / 256, 256, 0, stream>>>(h2, src, dst, nullptr, out);
    }
}